// BiAlignAggLayer_21818433864132
// MI455X (gfx1250) — compile-verified
//
#include <hip/hip_runtime.h>

#define BB 32
#define LL 1024
#define DD 128
#define HH 128
#define G3 384

typedef float v2f __attribute__((ext_vector_type(2)));
typedef float v8f __attribute__((ext_vector_type(8)));

__device__ __forceinline__ v8f wmma4(v2f a, v2f b, v8f c) {
  // D = A(16x4,f32) * B(4x16,f32) + C(16x16,f32), full f32 precision
  return __builtin_amdgcn_wmma_f32_16x16x4_f32(false, a, false, b, (short)0, c,
                                               false, false);
}

// CDNA5 async DMA: global -> LDS, 16B per op, tracked by ASYNCcnt.
typedef __attribute__((__vector_size__(4 * sizeof(int)))) int vi4;
typedef __attribute__((address_space(1))) vi4 gvi4;
typedef __attribute__((address_space(3))) vi4 svi4;

__device__ __forceinline__ void async_b128(void* lds, const void* g) {
  __builtin_amdgcn_global_load_async_to_lds_b128((gvi4*)g, (svi4*)lds, 0, 0);
}
__device__ __forceinline__ void wait_async0() {
  asm volatile("s_wait_asynccnt 0" ::: "memory");
}

// ---------------------------------------------------------------------------
// Kernel 1: S[b] = I[b] @ J[b]^T   (per batch 1024x1024, K=128)
// block = 256 thr (8 waves), block tile 128x128, wave tile 32x64 (2x4 tiles)
// ---------------------------------------------------------------------------
__global__ void k_gemm_s(const float* __restrict__ I, const float* __restrict__ J,
                         float* __restrict__ S) {
  extern __shared__ float sm[];
  float* As = sm;              // 128 x 132 (padded)
  float* Bs = sm + 128 * 132;  // 128 x 132
  const int b = blockIdx.z;
  const int row0 = blockIdx.x * 128;
  const int col0 = blockIdx.y * 128;
  const float* Ib = I + (size_t)b * LL * DD;
  const float* Jb = J + (size_t)b * LL * DD;
  const int t = threadIdx.x;

  for (int e = t; e < 128 * 32; e += 256) {  // 16B async DMA, full K resident
    int r = e >> 5, c4 = (e & 31) * 4;
    async_b128(As + r * 132 + c4, Ib + (size_t)(row0 + r) * DD + c4);
    async_b128(Bs + r * 132 + c4, Jb + (size_t)(col0 + r) * DD + c4);
  }
  wait_async0();
  __syncthreads();

  const int lane = t & 31, wid = t >> 5;
  const int l16 = lane & 15, lh2 = (lane >> 4) * 2, lh8 = (lane >> 4) * 8;
  const int wm = (wid & 3) * 32;   // 4 waves along M
  const int wn = (wid >> 2) * 64;  // 2 waves along N
  v8f acc[2][4] = {};

  for (int k = 0; k < 128; k += 4) {
    v2f a[2], bf[4];
#pragma unroll
    for (int im = 0; im < 2; im++) {
      const float* p = As + (wm + im * 16 + l16) * 132 + k + lh2;
      a[im] = (v2f){p[0], p[1]};
    }
#pragma unroll
    for (int in = 0; in < 4; in++) {
      const float* p = Bs + (wn + in * 16 + l16) * 132 + k + lh2;
      bf[in] = (v2f){p[0], p[1]};  // NT: B_w[k][n] = J[n][k]
    }
#pragma unroll
    for (int im = 0; im < 2; im++)
#pragma unroll
      for (int in = 0; in < 4; in++) acc[im][in] = wmma4(a[im], bf[in], acc[im][in]);
  }

  float* Sb = S + (size_t)b * LL * LL;
#pragma unroll
  for (int im = 0; im < 2; im++)
#pragma unroll
    for (int in = 0; in < 4; in++) {
      int gr0 = row0 + wm + im * 16 + lh8;
      int gc = col0 + wn + in * 16 + l16;
#pragma unroll
      for (int r = 0; r < 8; r++) Sb[(size_t)(gr0 + r) * LL + gc] = acc[im][in][r];
    }
}

// ---------------------------------------------------------------------------
// Kernel 2a: per-row (over m) max / sumexp.  One wave per row.
// ---------------------------------------------------------------------------
__global__ void k_rowstats(const float* __restrict__ S, float* __restrict__ rmax,
                           float* __restrict__ rsum) {
  const int row = blockIdx.x * 8 + (threadIdx.x >> 5);  // row in [0, B*L)
  const int lane = threadIdx.x & 31;
  const float* p = S + (size_t)row * LL;
  float m = -3.402823466e38f;
  for (int c = lane; c < LL; c += 32) m = fmaxf(m, p[c]);
  for (int off = 16; off; off >>= 1) m = fmaxf(m, __shfl_xor(m, off));
  float s = 0.f;
  for (int c = lane; c < LL; c += 32) s += __expf(p[c] - m);
  for (int off = 16; off; off >>= 1) s += __shfl_xor(s, off);
  if (lane == 0) { rmax[row] = m; rsum[row] = s; }
}

// ---------------------------------------------------------------------------
// Kernel 2b: per-column (over l) max / sumexp. thread = one column m.
// ---------------------------------------------------------------------------
__global__ void k_colstats(const float* __restrict__ S, float* __restrict__ cmax,
                           float* __restrict__ csum) {
  const int b = blockIdx.y;
  const int m = blockIdx.x * 256 + threadIdx.x;
  const float* Sb = S + (size_t)b * LL * LL;
  float mx = -3.402823466e38f;
  for (int l = 0; l < LL; l++) mx = fmaxf(mx, Sb[(size_t)l * LL + m]);
  float s = 0.f;
  for (int l = 0; l < LL; l++) s += __expf(Sb[(size_t)l * LL + m] - mx);
  cmax[b * LL + m] = mx;
  csum[b * LL + m] = s;
}

// ---------------------------------------------------------------------------
// Kernel 3: OUT[row,d] = | X[row,d] - sum_k P[row,k] * Y[k,d] |
//   mode 0 (out_i): P[l,m] = exp(S[l,m]-cmax[m])/csum[m], X=I, Y=J
//   mode 1 (out_j): P[m,l] = exp(S[l,m]-rmax[l])/rsum[l], X=J, Y=I (S read ^T)
// Async DMA of Y tile overlaps with the exp/normalize staging of P.
// ---------------------------------------------------------------------------
__global__ void k_wabs(const float* __restrict__ S, const float* __restrict__ stmax,
                       const float* __restrict__ stsum, const float* __restrict__ X,
                       const float* __restrict__ Y, float* __restrict__ OUT, int mode) {
  extern __shared__ float sm[];
  float* P = sm;             // 64 x 36 (padded)
  float* Yt = sm + 64 * 36;  // 32 x 132 (padded)
  const int b = blockIdx.y;
  const int row0 = blockIdx.x * 64;
  const float* Sb = S + (size_t)b * LL * LL;
  const float* Xb = X + (size_t)b * LL * DD;
  const float* Yb = Y + (size_t)b * LL * DD;
  const float* smax = stmax + b * LL;
  const float* ssum = stsum + b * LL;
  const int t = threadIdx.x, lane = t & 31, wid = t >> 5;
  const int l16 = lane & 15, lh2 = (lane >> 4) * 2, lh8 = (lane >> 4) * 8;
  const int wm = (wid & 3) * 16;   // 4 waves along M (1 tile each)
  const int wn = (wid >> 2) * 64;  // 2 waves along N (4 tiles each)
  v8f acc[4] = {};

  for (int k0 = 0; k0 < LL; k0 += 32) {
    // kick off the Y-tile DMA first, then compute softmax weights on the VALU
    for (int e = t; e < 32 * 32; e += 256) {
      int r = e >> 5, c4 = (e & 31) * 4;
      async_b128(Yt + r * 132 + c4, Yb + (size_t)(k0 + r) * DD + c4);
    }
    for (int e = t; e < 64 * 32; e += 256) {
      int r, c;
      if (mode == 0) { r = e >> 5; c = e & 31; }
      else           { r = e & 63; c = e >> 6; }
      float sv = (mode == 0) ? Sb[(size_t)(row0 + r) * LL + (k0 + c)]
                             : Sb[(size_t)(k0 + c) * LL + (row0 + r)];
      P[r * 36 + c] = __expf(sv - smax[k0 + c]) * (1.0f / ssum[k0 + c]);
    }
    wait_async0();
    __syncthreads();
#pragma unroll
    for (int kk = 0; kk < 32; kk += 4) {
      const float* pa = P + (wm + l16) * 36 + kk + lh2;
      v2f a = (v2f){pa[0], pa[1]};
#pragma unroll
      for (int in = 0; in < 4; in++) {
        const float* pb = Yt + (kk + lh2) * 132 + wn + in * 16 + l16;
        v2f bf = (v2f){pb[0], pb[132]};  // B_w[k][n] = Y[k][n], K-major
        acc[in] = wmma4(a, bf, acc[in]);
      }
    }
    __syncthreads();
  }

  float* Ob = OUT + (size_t)b * LL * DD;
#pragma unroll
  for (int in = 0; in < 4; in++) {
    int gc = wn + in * 16 + l16;
    int gr0 = row0 + wm + lh8;
#pragma unroll
    for (int r = 0; r < 8; r++) {
      float x = Xb[(size_t)(gr0 + r) * DD + gc];
      Ob[(size_t)(gr0 + r) * DD + gc] = fabsf(x - acc[in][r]);
    }
  }
}

// ---------------------------------------------------------------------------
// Kernel 4: xp[run] = src @ W + bias.  M=B*L=32768, K=128, N=384.
// grid.z = run in {i@Wf, i@Wb, j@Wf, j@Wb}; block tile 64x384, W in LDS.
// ---------------------------------------------------------------------------
__global__ void k_xp(const float* __restrict__ outI, const float* __restrict__ outJ,
                     const float* __restrict__ Wf, const float* __restrict__ Wb,
                     const float* __restrict__ bf_, const float* __restrict__ bb_,
                     float* __restrict__ xp) {
  extern __shared__ float sm[];
  float* Asm = sm;             // 64 x 132
  float* Wsm = sm + 64 * 132;  // 128 x 388
  const int run = blockIdx.z;
  const float* X = (run < 2) ? outI : outJ;
  const float* W = (run & 1) ? Wb : Wf;
  const float* bias = (run & 1) ? bb_ : bf_;
  float* XP = xp + (size_t)run * ((size_t)BB * LL * G3);
  const int row0 = blockIdx.x * 64;
  const int t = threadIdx.x;

  for (int e = t; e < 64 * 32; e += 256) {
    int r = e >> 5, c4 = (e & 31) * 4;
    async_b128(Asm + r * 132 + c4, X + (size_t)(row0 + r) * DD + c4);
  }
  for (int e = t; e < 128 * 96; e += 256) {
    int r = e / 96, c4 = (e % 96) * 4;
    async_b128(Wsm + r * 388 + c4, W + (size_t)r * G3 + c4);
  }
  wait_async0();
  __syncthreads();

  const int lane = t & 31, wid = t >> 5;
  const int l16 = lane & 15, lh2 = (lane >> 4) * 2, lh8 = (lane >> 4) * 8;
  const int wm = (wid & 1) * 32;   // 2 waves along M (2 tiles each)
  const int wn = (wid >> 1) * 96;  // 4 waves along N (6 tiles each)
  v8f acc[2][6] = {};

  for (int k = 0; k < 128; k += 4) {
    v2f a[2];
#pragma unroll
    for (int im = 0; im < 2; im++) {
      const float* p = Asm + (wm + im * 16 + l16) * 132 + k + lh2;
      a[im] = (v2f){p[0], p[1]};
    }
#pragma unroll
    for (int in = 0; in < 6; in++) {
      const float* p = Wsm + (k + lh2) * 388 + wn + in * 16 + l16;
      v2f bf = (v2f){p[0], p[388]};
#pragma unroll
      for (int im = 0; im < 2; im++) acc[im][in] = wmma4(a[im], bf, acc[im][in]);
    }
  }

#pragma unroll
  for (int im = 0; im < 2; im++)
#pragma unroll
    for (int in = 0; in < 6; in++) {
      int gc = wn + in * 16 + l16;
      int gr0 = row0 + wm + im * 16 + lh8;
#pragma unroll
      for (int r = 0; r < 8; r++)
        XP[(size_t)(gr0 + r) * G3 + gc] = acc[im][in][r] + bias[gc];
    }
}

// ---------------------------------------------------------------------------
// Kernel 5: GRU recurrence. grid=4 (run = {i,f},{i,b},{j,f},{j,b}), block=256.
// U (192KB) + h (16.5KB) + gate preacts (48.5KB) live in the 320KB WGP LDS.
// Per step: a = h @ U via WMMA (2x24 tiles over 8 waves), fused gate math.
// ---------------------------------------------------------------------------
__global__ void k_gru(const float* __restrict__ xp, const float* __restrict__ Uf,
                      const float* __restrict__ Ub, float* __restrict__ hfin) {
  extern __shared__ float sm[];
  float* Ush = sm;                    // 128 x 388
  float* hsh = Ush + 128 * 388;       // 32 x 132
  float* ash = hsh + 32 * 132;        // 32 x 388
  const int run = blockIdx.x;
  const float* U = (run & 1) ? Ub : Uf;
  const float* XP = xp + (size_t)run * ((size_t)BB * LL * G3);
  const int fwd = !(run & 1);
  const int t = threadIdx.x;

  for (int e = t; e < 128 * 96; e += 256) {
    int r = e / 96, c4 = (e % 96) * 4;
    async_b128(Ush + r * 388 + c4, U + (size_t)r * G3 + c4);
  }
  for (int e = t; e < 32 * 132; e += 256) hsh[e] = 0.f;
  wait_async0();
  __syncthreads();

  const int lane = t & 31, wid = t >> 5;
  const int l16 = lane & 15, lh2 = (lane >> 4) * 2, lh8 = (lane >> 4) * 8;
  const int wm = (wid & 1) * 16;   // 2 M tiles across waves
  const int wn = (wid >> 1) * 96;  // 4 N groups x 6 tiles

  for (int step = 0; step < LL; step++) {
    const int tt = fwd ? step : (LL - 1 - step);
    v8f acc[6] = {};
    for (int k = 0; k < 128; k += 4) {
      const float* ph = hsh + (wm + l16) * 132 + k + lh2;
      v2f a = (v2f){ph[0], ph[1]};
#pragma unroll
      for (int in = 0; in < 6; in++) {
        const float* p = Ush + (k + lh2) * 388 + wn + in * 16 + l16;
        v2f bf = (v2f){p[0], p[388]};
        acc[in] = wmma4(a, bf, acc[in]);
      }
    }
#pragma unroll
    for (int in = 0; in < 6; in++) {
      int gc = wn + in * 16 + l16;
#pragma unroll
      for (int r = 0; r < 8; r++) ash[(wm + lh8 + r) * 388 + gc] = acc[in][r];
    }
    // prefetch next step's xp slice into L2 while gates compute
    if (step + 1 < LL && t < BB) {
      const int tn = fwd ? (step + 1) : (LL - 2 - step);
      __builtin_prefetch(XP + ((size_t)t * LL + tn) * G3, 0, 1);
    }
    __syncthreads();
    for (int e = t; e < BB * HH; e += 256) {
      int m = e >> 7, n = e & 127;
      const float* xrow = XP + ((size_t)m * LL + tt) * G3;
      float xz = xrow[n], xr = xrow[n + 128], xh = xrow[n + 256];
      float az = ash[m * 388 + n], ar = ash[m * 388 + n + 128], ah = ash[m * 388 + n + 256];
      float z = 1.f / (1.f + __expf(-(xz + az)));
      float r = 1.f / (1.f + __expf(-(xr + ar)));
      float hh = tanhf(xh + r * ah);
      float hp = hsh[m * 132 + n];
      hsh[m * 132 + n] = z * hp + (1.f - z) * hh;
    }
    __syncthreads();
  }

  for (int e = t; e < BB * HH; e += 256) {
    int m = e >> 7, n = e & 127;
    hfin[(size_t)run * BB * HH + e] = hsh[m * 132 + n];
  }
}

// ---------------------------------------------------------------------------
// Kernel 6: out = 0.5*(concat(h0,h1) + concat(h2,h3))
// ---------------------------------------------------------------------------
__global__ void k_final(const float* __restrict__ hfin, float* __restrict__ out) {
  const int e = blockIdx.x * 256 + threadIdx.x;
  if (e >= BB * 2 * HH) return;
  const int b = e >> 8, c = e & 255;
  float v;
  if (c < HH)
    v = 0.5f * (hfin[0 * BB * HH + b * HH + c] + hfin[2 * BB * HH + b * HH + c]);
  else
    v = 0.5f * (hfin[1 * BB * HH + b * HH + (c - HH)] + hfin[3 * BB * HH + b * HH + (c - HH)]);
  out[e] = v;
}

// ---------------------------------------------------------------------------
extern "C" void kernel_launch(void* const* d_in, const int* in_sizes, int n_in,
                              void* d_out, int out_size, void* d_ws, size_t ws_size,
                              hipStream_t stream) {
  const float* I  = (const float*)d_in[0];
  const float* J  = (const float*)d_in[1];
  const float* Wf = (const float*)d_in[2];
  const float* Uf = (const float*)d_in[3];
  const float* bf = (const float*)d_in[4];
  const float* Wb = (const float*)d_in[5];
  const float* Ub = (const float*)d_in[6];
  const float* bb = (const float*)d_in[7];
  float* out = (float*)d_out;

  float* ws = (float*)d_ws;
  size_t off = 0;
  float* S    = ws + off; off += (size_t)BB * LL * LL;
  float* rmax = ws + off; off += (size_t)BB * LL;
  float* rsum = ws + off; off += (size_t)BB * LL;
  float* cmax = ws + off; off += (size_t)BB * LL;
  float* csum = ws + off; off += (size_t)BB * LL;
  float* outI = ws + off; off += (size_t)BB * LL * DD;
  float* outJ = ws + off; off += (size_t)BB * LL * DD;
  float* xp   = ws + off; off += (size_t)4 * BB * LL * G3;
  float* hfin = ws + off; off += (size_t)4 * BB * HH;

  const size_t lds_gemm = (size_t)(2 * 128 * 132) * sizeof(float);
  const size_t lds_wabs = (size_t)(64 * 36 + 32 * 132) * sizeof(float);
  const size_t lds_xp   = (size_t)(64 * 132 + 128 * 388) * sizeof(float);
  const size_t lds_gru  = (size_t)(128 * 388 + 32 * 132 + 32 * 388) * sizeof(float);
  (void)hipFuncSetAttribute((const void*)k_gemm_s, hipFuncAttributeMaxDynamicSharedMemorySize, (int)lds_gemm);
  (void)hipFuncSetAttribute((const void*)k_wabs,   hipFuncAttributeMaxDynamicSharedMemorySize, (int)lds_wabs);
  (void)hipFuncSetAttribute((const void*)k_xp,     hipFuncAttributeMaxDynamicSharedMemorySize, (int)lds_xp);
  (void)hipFuncSetAttribute((const void*)k_gru,    hipFuncAttributeMaxDynamicSharedMemorySize, (int)lds_gru);

  k_gemm_s<<<dim3(8, 8, BB), 256, lds_gemm, stream>>>(I, J, S);
  k_rowstats<<<BB * LL / 8, 256, 0, stream>>>(S, rmax, rsum);
  k_colstats<<<dim3(LL / 256, BB), 256, 0, stream>>>(S, cmax, csum);
  k_wabs<<<dim3(LL / 64, BB), 256, lds_wabs, stream>>>(S, cmax, csum, I, J, outI, 0);
  k_wabs<<<dim3(LL / 64, BB), 256, lds_wabs, stream>>>(S, rmax, rsum, J, I, outJ, 1);
  k_xp<<<dim3(BB * LL / 64, 1, 4), 256, lds_xp, stream>>>(outI, outJ, Wf, Wb, bf, bb, xp);
  k_gru<<<4, 256, lds_gru, stream>>>(xp, Uf, Ub, hfin);
  k_final<<<(BB * 2 * HH + 255) / 256, 256, 0, stream>>>(hfin, out);
}